// OSAMCrossBlock_71433896067311
// MI455X (gfx1250) — compile-verified
//
#include <hip/hip_runtime.h>
#include <math.h>

#define BB 8
#define TT 128
#define DD 512
#define MM 2048
#define HH 128

typedef float v2f __attribute__((ext_vector_type(2)));
typedef float v8f __attribute__((ext_vector_type(8)));

__device__ __forceinline__ v8f wmma_f32(v2f a, v2f b, v8f c) {
    // D = A(16x4) * B(4x16) + C(16x16), f32, wave32
    return __builtin_amdgcn_wmma_f32_16x16x4_f32(false, a, false, b, (short)0, c, false, false);
}

__device__ __forceinline__ float warp_sum(float x) {
#pragma unroll
    for (int m = 16; m; m >>= 1) x += __shfl_xor(x, m, 32);
    return x;
}

__device__ __forceinline__ float gelu_exact(float x) {
    return 0.5f * x * (1.0f + erff(x * 0.70710678118654752f));
}
__device__ __forceinline__ float sigmoidf(float x) { return 1.0f / (1.0f + __expf(-x)); }

// ---------------------------------------------------------------------------
// Kernel A: q/k/v GEMMs (WMMA f32) + count-sketch scatter (LDS atomics)
// ---------------------------------------------------------------------------
__global__ __launch_bounds__(256)
void k_front(const float* __restrict__ e, const int* __restrict__ h,
             const float* __restrict__ sign,
             const float* __restrict__ Wq, const float* __restrict__ Wk,
             const float* __restrict__ Wv,
             const float* __restrict__ thq_raw, const float* __restrict__ thk_raw,
             const float* __restrict__ om_raw,
             float* __restrict__ q, float* __restrict__ k, float* __restrict__ v,
             float* __restrict__ Qre, float* __restrict__ Qim,
             float* __restrict__ dKre, float* __restrict__ dKim, int t)
{
    __shared__ float smem[16 * DD];  // 32 KB; reused: e-tile then 4x2048 sketch bins
    const int tid = threadIdx.x;
    const int lane = tid & 31, wave = tid >> 5;

    // Load e_t (8xD) zero-padded to 16 rows (WMMA A tile source)
    for (int i = tid; i < 16 * DD; i += 256) {
        int m = i >> 9, d = i & (DD - 1);
        smem[i] = (m < BB) ? e[((size_t)m * TT + t) * DD + d] : 0.0f;
    }
    __syncthreads();

    const int mrow = lane & 15;          // A: M = lane%16
    const int khi  = (lane >> 4) * 2;    // A/B: K pair select by lane half
    for (int job = wave; job < 96; job += 8) {
        const int which = job >> 5, ntile = job & 31;
        const float* W = (which == 0) ? Wq : (which == 1) ? Wk : Wv;
        float* out = (which == 0) ? q : (which == 1) ? k : v;
        const int n = ntile * 16 + (lane & 15);
        v8f acc = {0.f, 0.f, 0.f, 0.f, 0.f, 0.f, 0.f, 0.f};
        for (int k0 = 0; k0 < DD; k0 += 4) {
            v2f av; av.x = smem[mrow * DD + k0 + khi];
                    av.y = smem[mrow * DD + k0 + khi + 1];
            v2f bv; bv.x = W[(size_t)(k0 + khi) * DD + n];
                    bv.y = W[(size_t)(k0 + khi + 1) * DD + n];
            acc = wmma_f32(av, bv, acc);
        }
        if (lane < 16) {  // C/D: VGPR r -> row r for lanes 0..15; rows 0..7 are real batches
#pragma unroll
            for (int r = 0; r < 8; ++r) out[r * DD + n] = acc[r];
        }
    }
    __threadfence();
    __syncthreads();

    // Count-sketch scatter per batch into LDS bins
    const float thq = 0.5f * tanhf(thq_raw[0]);
    const float thk = 0.5f * tanhf(thk_raw[0]);
    const float om  = 0.5f * tanhf(om_raw[0]);
    const float posf = (float)t;
    float* sQre = smem;           float* sQim = smem + MM;
    float* sKre = smem + 2 * MM;  float* sKim = smem + 3 * MM;
    for (int b = 0; b < BB; ++b) {
        for (int i = tid; i < 4 * MM; i += 256) smem[i] = 0.0f;
        __syncthreads();
        for (int d = tid; d < DD; d += 256) {
            const int m = h[d];
            const float sg = sign[d];
            const float qv = q[b * DD + d], kv = k[b * DD + d];
            const float phq = thq * qv - om * posf;
            const float phk = thk * kv + om * posf;
            float cq, sq, ck, sk;
            __sincosf(phq, &sq, &cq);
            __sincosf(phk, &sk, &ck);
            atomicAdd(&sQre[m], qv * cq * sg);
            atomicAdd(&sQim[m], qv * sq * sg);
            atomicAdd(&sKre[m], kv * ck * sg);
            atomicAdd(&sKim[m], kv * sk * sg);
        }
        __syncthreads();
        for (int i = tid; i < MM; i += 256) {
            Qre[b * MM + i]  = sQre[i];  Qim[b * MM + i]  = sQim[i];
            dKre[b * MM + i] = sKre[i];  dKim[b * MM + i] = sKim[i];
        }
        __syncthreads();
    }
}

// ---------------------------------------------------------------------------
// Kernel F: per-batch 2048-pt complex FFT conv (LDS), psi, K update in-place
// ---------------------------------------------------------------------------
__device__ void fft2048(float* re, float* im, int tid, float sgn) {
    // bit-reversal permutation (11 bits)
    for (int i = tid; i < MM; i += 256) {
        int j = (int)(__brev((unsigned)i) >> 21);
        if (j > i) {
            float tr = re[i]; re[i] = re[j]; re[j] = tr;
            float ti = im[i]; im[i] = im[j]; im[j] = ti;
        }
    }
    __syncthreads();
    for (int len = 2; len <= MM; len <<= 1) {
        const int half = len >> 1;
        const float ang = sgn * 6.283185307179586f / (float)len;
        for (int kk = tid; kk < MM / 2; kk += 256) {
            const int blk = kk / half;
            const int off = kk - blk * half;
            const int i0 = blk * len + off, i1 = i0 + half;
            float s, c;
            __sincosf(ang * (float)off, &s, &c);
            const float xr = re[i1], xi = im[i1];
            const float tr = xr * c - xi * s;
            const float ti = xr * s + xi * c;
            re[i1] = re[i0] - tr; im[i1] = im[i0] - ti;
            re[i0] += tr;         im[i0] += ti;
        }
        __syncthreads();
    }
    const float s = 0.02209708691207961f;  // 1/sqrt(2048), 'ortho'
    for (int i = tid; i < MM; i += 256) { re[i] *= s; im[i] *= s; }
    __syncthreads();
}

__global__ __launch_bounds__(256)
void k_fft(const float* __restrict__ Qre, const float* __restrict__ Qim,
           const float* __restrict__ dKre, const float* __restrict__ dKim,
           float* __restrict__ Kst, float* __restrict__ psi, int t)
{
    __shared__ float qre[MM], qim[MM], kre[MM], kim[MM];  // 32 KB
    const int tid = threadIdx.x;
    const int b = blockIdx.x;
    for (int i = tid; i < MM; i += 256) {
        qre[i] = Qre[b * MM + i]; qim[i] = Qim[b * MM + i];
        kre[i] = Kst[(b * 2 + 0) * MM + i];  // old K (pre-update), as in reference
        kim[i] = Kst[(b * 2 + 1) * MM + i];
    }
    __syncthreads();
    fft2048(qre, qim, tid, -1.0f);
    fft2048(kre, kim, tid, -1.0f);
    for (int i = tid; i < MM; i += 256) {
        const float ar = qre[i], ai = qim[i], br = kre[i], bi = kim[i];
        qre[i] = ar * br - ai * bi;
        qim[i] = ar * bi + ai * br;
    }
    __syncthreads();
    fft2048(qre, qim, tid, +1.0f);
    const float its = rsqrtf((float)t + 1.0f);
    for (int i = tid; i < MM; i += 256) {
        psi[b * MM + i] = qre[i] * its;
        Kst[(b * 2 + 0) * MM + i] += dKre[b * MM + i];
        Kst[(b * 2 + 1) * MM + i] += dKim[b * MM + i];
    }
}

// ---------------------------------------------------------------------------
// Kernel R: h_feat = gelu(psi @ ro_w1 + b1) @ ro_w2 + b2   (WMMA f32)
// ---------------------------------------------------------------------------
__global__ __launch_bounds__(256)
void k_readout(const float* __restrict__ psi, const float* __restrict__ w1,
               const float* __restrict__ b1, const float* __restrict__ w2,
               const float* __restrict__ b2, float* __restrict__ hfeat)
{
    __shared__ float hid[16 * HH];  // 8 KB
    const int tid = threadIdx.x, lane = tid & 31, wave = tid >> 5;
    const int mrow = lane & 15, khi = (lane >> 4) * 2;
    {
        const int n = wave * 16 + (lane & 15);  // 8 waves cover N=128
        v8f acc = {0.f, 0.f, 0.f, 0.f, 0.f, 0.f, 0.f, 0.f};
        for (int k0 = 0; k0 < MM; k0 += 4) {
            v2f av;
            av.x = (mrow < BB) ? psi[mrow * MM + k0 + khi] : 0.0f;
            av.y = (mrow < BB) ? psi[mrow * MM + k0 + khi + 1] : 0.0f;
            v2f bv; bv.x = w1[(size_t)(k0 + khi) * HH + n];
                    bv.y = w1[(size_t)(k0 + khi + 1) * HH + n];
            acc = wmma_f32(av, bv, acc);
        }
        const int mbase = (lane >> 4) * 8;
#pragma unroll
        for (int r = 0; r < 8; ++r)
            hid[(mbase + r) * HH + n] = gelu_exact(acc[r] + b1[n]);
    }
    __syncthreads();
    for (int job = wave; job < 32; job += 8) {
        const int n = job * 16 + (lane & 15);
        v8f acc = {0.f, 0.f, 0.f, 0.f, 0.f, 0.f, 0.f, 0.f};
        for (int k0 = 0; k0 < HH; k0 += 4) {
            v2f av; av.x = hid[mrow * HH + k0 + khi];
                    av.y = hid[mrow * HH + k0 + khi + 1];
            v2f bv; bv.x = w2[(size_t)(k0 + khi) * DD + n];
                    bv.y = w2[(size_t)(k0 + khi + 1) * DD + n];
            acc = wmma_f32(av, bv, acc);
        }
        if (lane < 16) {
#pragma unroll
            for (int r = 0; r < 8; ++r) hfeat[r * DD + n] = acc[r] + b2[n];
        }
    }
}

// ---------------------------------------------------------------------------
// Kernel B (dominant): fused vread contraction + rank-1 H update.
// One wave per (b,d): 2x 8KB contiguous H rows, L2-resident. float4 + prefetch.
// ---------------------------------------------------------------------------
__global__ __launch_bounds__(256)
void k_state(float* __restrict__ H, const float* __restrict__ Qre,
             const float* __restrict__ Qim, const float* __restrict__ dKre,
             const float* __restrict__ dKim, const float* __restrict__ v,
             float* __restrict__ vread, int t)
{
    const int lane = threadIdx.x & 31;
    const int wid = (int)((blockIdx.x * blockDim.x + threadIdx.x) >> 5);  // 0..4095
    const int b = wid >> 9;
    const int d = wid & (DD - 1);
    const float vb = v[b * DD + d];
    float4* H0 = (float4*)(H + ((size_t)(b * 2 + 0) * DD + d) * MM);
    float4* H1 = (float4*)(H + ((size_t)(b * 2 + 1) * DD + d) * MM);
    const float4* qr = (const float4*)(Qre + b * MM);
    const float4* qi = (const float4*)(Qim + b * MM);
    const float4* kr = (const float4*)(dKre + b * MM);
    const float4* ki = (const float4*)(dKim + b * MM);
    float acc = 0.0f;
#pragma unroll 4
    for (int it = 0; it < MM / 128; ++it) {
        const int idx = it * 32 + lane;
        if (it + 2 < MM / 128) {
            __builtin_prefetch(&H0[(it + 2) * 32 + lane], 1, 0);  // global_prefetch
            __builtin_prefetch(&H1[(it + 2) * 32 + lane], 1, 0);
        }
        float4 h0 = H0[idx], h1 = H1[idx];
        const float4 a = qr[idx], bq = qi[idx], cr = kr[idx], ci = ki[idx];
        acc += h0.x * a.x + h0.y * a.y + h0.z * a.z + h0.w * a.w;
        acc -= h1.x * bq.x + h1.y * bq.y + h1.z * bq.z + h1.w * bq.w;
        h0.x += vb * cr.x; h0.y += vb * cr.y; h0.z += vb * cr.z; h0.w += vb * cr.w;
        h1.x += vb * ci.x; h1.y += vb * ci.y; h1.z += vb * ci.z; h1.w += vb * ci.w;
        H0[idx] = h0; H1[idx] = h1;
    }
    acc = warp_sum(acc);
    if (lane == 0) vread[b * DD + d] = acc * rsqrtf((float)t + 1.0f);
}

// ---------------------------------------------------------------------------
// Kernel V: LN(vread) -> vp MLP (WMMA f32) -> residual combine -> final LN -> y
// ---------------------------------------------------------------------------
__global__ __launch_bounds__(256)
void k_value(const float* __restrict__ e, const float* __restrict__ vread,
             const float* __restrict__ hfeat,
             const float* __restrict__ vp_ln_g, const float* __restrict__ vp_ln_b,
             const float* __restrict__ vp_w1, const float* __restrict__ vp_b1,
             const float* __restrict__ vp_w2, const float* __restrict__ vp_b2,
             const float* __restrict__ ln_g, const float* __restrict__ ln_b,
             const float* __restrict__ gf_raw, const float* __restrict__ gv_raw,
             float* __restrict__ ys, int t)
{
    __shared__ float A[16 * DD];   // 32 KB
    __shared__ float Hd[16 * DD];  // 32 KB
    const int tid = threadIdx.x, lane = tid & 31, wave = tid >> 5;
    const int mrow = lane & 15, khi = (lane >> 4) * 2;

    if (wave < BB) {  // LayerNorm(vread) rows -> A
        const int b = wave;
        float s = 0.f, s2 = 0.f;
        for (int d = lane; d < DD; d += 32) { const float x = vread[b * DD + d]; s += x; s2 += x * x; }
        s = warp_sum(s); s2 = warp_sum(s2);
        const float mu = s * (1.0f / DD);
        const float inv = rsqrtf(s2 * (1.0f / DD) - mu * mu + 1e-5f);
        for (int d = lane; d < DD; d += 32) {
            const float x = vread[b * DD + d];
            A[b * DD + d] = (x - mu) * inv * vp_ln_g[d] + vp_ln_b[d];
        }
    }
    for (int i = tid; i < 8 * DD; i += 256) A[8 * DD + i] = 0.0f;  // zero pad rows
    __syncthreads();

    for (int job = wave; job < 32; job += 8) {  // @ vp_w1, gelu -> Hd
        const int n = job * 16 + (lane & 15);
        v8f acc = {0.f, 0.f, 0.f, 0.f, 0.f, 0.f, 0.f, 0.f};
        for (int k0 = 0; k0 < DD; k0 += 4) {
            v2f av; av.x = A[mrow * DD + k0 + khi];
                    av.y = A[mrow * DD + k0 + khi + 1];
            v2f bv; bv.x = vp_w1[(size_t)(k0 + khi) * DD + n];
                    bv.y = vp_w1[(size_t)(k0 + khi + 1) * DD + n];
            acc = wmma_f32(av, bv, acc);
        }
        const int mbase = (lane >> 4) * 8;
#pragma unroll
        for (int r = 0; r < 8; ++r)
            Hd[(mbase + r) * DD + n] = gelu_exact(acc[r] + vp_b1[n]);
    }
    __syncthreads();

    const float gf = 0.25f * sigmoidf(gf_raw[0]);
    const float gv = 0.25f * sigmoidf(gv_raw[0]);
    for (int job = wave; job < 32; job += 8) {  // @ vp_w2, residual combine -> A
        const int n = job * 16 + (lane & 15);
        v8f acc = {0.f, 0.f, 0.f, 0.f, 0.f, 0.f, 0.f, 0.f};
        for (int k0 = 0; k0 < DD; k0 += 4) {
            v2f av; av.x = Hd[mrow * DD + k0 + khi];
                    av.y = Hd[mrow * DD + k0 + khi + 1];
            v2f bv; bv.x = vp_w2[(size_t)(k0 + khi) * DD + n];
                    bv.y = vp_w2[(size_t)(k0 + khi + 1) * DD + n];
            acc = wmma_f32(av, bv, acc);
        }
        if (lane < 16) {
#pragma unroll
            for (int r = 0; r < 8; ++r) {
                const float vm = acc[r] + vp_b2[n];
                A[r * DD + n] = e[((size_t)r * TT + t) * DD + n]
                              + gf * hfeat[r * DD + n] + gv * vm;
            }
        }
    }
    __syncthreads();

    if (wave < BB) {  // final LayerNorm -> ys[:, t, :]
        const int b = wave;
        float s = 0.f, s2 = 0.f;
        for (int d = lane; d < DD; d += 32) { const float x = A[b * DD + d]; s += x; s2 += x * x; }
        s = warp_sum(s); s2 = warp_sum(s2);
        const float mu = s * (1.0f / DD);
        const float inv = rsqrtf(s2 * (1.0f / DD) - mu * mu + 1e-5f);
        for (int d = lane; d < DD; d += 32)
            ys[((size_t)b * TT + t) * DD + d] = (A[b * DD + d] - mu) * inv * ln_g[d] + ln_b[d];
    }
}

// ---------------------------------------------------------------------------
extern "C" void kernel_launch(void* const* d_in, const int* in_sizes, int n_in,
                              void* d_out, int out_size, void* d_ws, size_t ws_size,
                              hipStream_t stream)
{
    const float* e      = (const float*)d_in[0];
    const int*   h      = (const int*)d_in[1];
    const float* sgn    = (const float*)d_in[2];
    const float* Wq     = (const float*)d_in[3];
    const float* Wk     = (const float*)d_in[4];
    const float* Wv     = (const float*)d_in[5];
    const float* ro_w1  = (const float*)d_in[6];
    const float* ro_b1  = (const float*)d_in[7];
    const float* ro_w2  = (const float*)d_in[8];
    const float* ro_b2  = (const float*)d_in[9];
    const float* vpln_g = (const float*)d_in[10];
    const float* vpln_b = (const float*)d_in[11];
    const float* vp_w1  = (const float*)d_in[12];
    const float* vp_b1  = (const float*)d_in[13];
    const float* vp_w2  = (const float*)d_in[14];
    const float* vp_b2  = (const float*)d_in[15];
    const float* ln_g   = (const float*)d_in[16];
    const float* ln_b   = (const float*)d_in[17];
    const float* thq    = (const float*)d_in[18];
    const float* thk    = (const float*)d_in[19];
    const float* om     = (const float*)d_in[20];
    const float* gf     = (const float*)d_in[21];
    const float* gv     = (const float*)d_in[22];

    float* ws    = (float*)d_ws;
    float* q     = ws;
    float* k     = q + BB * DD;
    float* v     = k + BB * DD;
    float* Qre   = v + BB * DD;
    float* Qim   = Qre + BB * MM;
    float* dKre  = Qim + BB * MM;
    float* dKim  = dKre + BB * MM;
    float* psi   = dKim + BB * MM;
    float* hfeat = psi + BB * MM;
    float* vread = hfeat + BB * DD;

    float* out = (float*)d_out;
    float* ys  = out;
    float* Kst = out + (size_t)BB * TT * DD;              // Kf: state held in-place
    float* Hst = Kst + (size_t)BB * 2 * MM;               // Hf: state held in-place

    // Zero K/H state once (capture-safe memset node)
    hipMemsetAsync(Kst, 0,
                   ((size_t)BB * 2 * MM + (size_t)BB * 2 * DD * MM) * sizeof(float),
                   stream);

    for (int t = 0; t < TT; ++t) {
        k_front  <<<1,   256, 0, stream>>>(e, h, sgn, Wq, Wk, Wv, thq, thk, om,
                                           q, k, v, Qre, Qim, dKre, dKim, t);
        k_fft    <<<BB,  256, 0, stream>>>(Qre, Qim, dKre, dKim, Kst, psi, t);
        k_readout<<<1,   256, 0, stream>>>(psi, ro_w1, ro_b1, ro_w2, ro_b2, hfeat);
        k_state  <<<512, 256, 0, stream>>>(Hst, Qre, Qim, dKre, dKim, v, vread, t);
        k_value  <<<1,   256, 0, stream>>>(e, vread, hfeat, vpln_g, vpln_b,
                                           vp_w1, vp_b1, vp_w2, vp_b2,
                                           ln_g, ln_b, gf, gv, ys, t);
    }
    (void)in_sizes; (void)n_in; (void)out_size; (void)ws_size;
}